// SNNEventModel_45363444580787
// MI455X (gfx1250) — compile-verified
//
#include <hip/hip_runtime.h>
#include <cmath>

// Problem constants (fixed shapes from the reference)
#define B_N     32
#define F_N     64
#define L_LEN   8192
#define K_TAPS  8

#define X_SCALE 20.0f
#define THR     0.25f
#define BETA    15.0f

typedef __attribute__((ext_vector_type(2))) float v2f;
typedef __attribute__((ext_vector_type(8))) float v8f;

// ---------------------------------------------------------------------------
// Stage 0: normalize filters to unit norm; alpha = exp(-1/(softplus(tau)+1e-4))
// ---------------------------------------------------------------------------
__global__ void snn_prep_kernel(const float* __restrict__ conv_w,
                                const float* __restrict__ raw_tau,
                                float* __restrict__ wn,
                                float* __restrict__ alpha) {
    int f = threadIdx.x;
    if (f >= F_N) return;
    float w[K_TAPS];
    float ss = 0.f;
    #pragma unroll
    for (int k = 0; k < K_TAPS; ++k) { w[k] = conv_w[f * K_TAPS + k]; ss += w[k] * w[k]; }
    float nrm = fmaxf(sqrtf(ss), 1e-8f);
    #pragma unroll
    for (int k = 0; k < K_TAPS; ++k) wn[f * K_TAPS + k] = w[k] / nrm;
    float r  = raw_tau[f];
    float sp = (r > 20.f) ? r : log1pf(expf(r));   // stable softplus
    alpha[f] = expf(-1.0f / (sp + 1e-4f));
}

// ---------------------------------------------------------------------------
// Stage 1: causal conv as WMMA f32 GEMM tiles.
//   D(16f x 16l) = A(16f x 8k) * B(8k x 16l), split K=8 into 2 x (K=4) WMMAs.
// ISA 32-bit A 16x4 layout: lanes 0-15 M=0..15; VGPR0 holds K=0 (lanes 0-15)
// and K=2 (lanes 16-31); VGPR1 holds K=1 / K=3. B (4x16) mirrors this with
// N striped across lanes. C/D 16x16: VGPR r -> M=r (lanes 0-15) / M=r+8.
// One wave32 per tile group; EXEC is all-ones at the WMMA (no divergence).
// ---------------------------------------------------------------------------
#define LT_PER_WAVE 8

__global__ void snn_conv_wmma_kernel(const float* __restrict__ x,
                                     const float* __restrict__ wn,
                                     float* __restrict__ I) {
    const int lane   = threadIdx.x;      // 0..31
    const int lchunk = blockIdx.x;       // 0..(512/LT_PER_WAVE - 1)
    const int ftile  = blockIdx.y;       // 0..3
    const int b      = blockIdx.z;       // 0..31

    const int n     = lane & 15;         // N index (l within tile) for B/C/D
    const int khalf = lane >> 4;         // 0: K pair {0,1}; 1: K pair {2,3}

    // A operands: constant across all l-tiles of this wave.
    v2f a0, a1;
    {
        const int fa = ftile * 16 + (lane & 15);   // M = lane&15 for both halves
        const int ka = 2 * khalf;
        a0.x = wn[fa * K_TAPS + ka];
        a0.y = wn[fa * K_TAPS + ka + 1];
        a1.x = wn[fa * K_TAPS + 4 + ka];
        a1.y = wn[fa * K_TAPS + 4 + ka + 1];
    }

    const float* xb = x + (size_t)b * L_LEN;

    for (int lt = 0; lt < LT_PER_WAVE; ++lt) {
        const int l0 = (lchunk * LT_PER_WAVE + lt) * 16;
        const int kb = 2 * khalf;

        // B[k][n] = X_SCALE * x[b, l0 + n + k - 7], zero for t < 0 (causal pad).
        int t;
        v2f b0, b1;
        t = l0 + n + kb - 7;         b0.x = (t >= 0) ? X_SCALE * xb[t] : 0.0f;
        t = l0 + n + kb - 6;         b0.y = (t >= 0) ? X_SCALE * xb[t] : 0.0f;
        t = l0 + n + 4 + kb - 7;     b1.x = (t >= 0) ? X_SCALE * xb[t] : 0.0f;
        t = l0 + n + 4 + kb - 6;     b1.y = (t >= 0) ? X_SCALE * xb[t] : 0.0f;

        v8f c = {};
        c = __builtin_amdgcn_wmma_f32_16x16x4_f32(false, a0, false, b0,
                                                  (short)0, c, false, false);
        c = __builtin_amdgcn_wmma_f32_16x16x4_f32(false, a1, false, b1,
                                                  (short)0, c, false, false);

        // Store D: row r -> f = ftile*16 + r + 8*khalf, col = l0 + n.
        float* Ibase = I + ((size_t)b * F_N + ftile * 16 + khalf * 8) * L_LEN + l0 + n;
        #pragma unroll
        for (int r = 0; r < 8; ++r)
            Ibase[(size_t)r * L_LEN] = c[r];
    }
}

// ---------------------------------------------------------------------------
// Stage 2: sequential LIF scan over L with double-buffered async global->LDS
// copies (gfx1250 GLOBAL_LOAD_ASYNC_TO_LDS_B128, tracked by ASYNCcnt), fused
// logits (max over F). One 64-thread block (2 waves) per batch; thread=filter.
// 65-float LDS row stride => bank-conflict-free for both the row-major
// cooperative phases and the column-wise compute phase.
// ---------------------------------------------------------------------------
#define TS      64            // timesteps per chunk
#define ROWPAD  65            // LDS row stride (floats)
#define NCHUNK  (L_LEN / TS)  // 128

__device__ __forceinline__ void async_tile_issue(const float* __restrict__ gsrc,
                                                 float* __restrict__ ldst_tile,
                                                 int tid, int t0) {
    // Each thread issues 16 x b128 async copies: tile row fr, 4-float column
    // group tc, fully coalesced on the global side.
    #pragma unroll
    for (int i = 0; i < 16; ++i) {
        int e  = i * 64 + tid;        // 0..1023 float4 slots
        int fr = e >> 4;              // row (filter)
        int tc = (e & 15) * 4;        // column (time)
        uint64_t ga = (uint64_t)(uintptr_t)(gsrc + (size_t)fr * L_LEN + t0 + tc);
        uint32_t la = (uint32_t)(uintptr_t)(&ldst_tile[fr * ROWPAD + tc]);
        asm volatile("global_load_async_to_lds_b128 %0, %1, off"
                     :: "v"(la), "v"(ga) : "memory");
    }
}

__global__ void snn_scan_kernel(const float* __restrict__ Iin,
                                const float* __restrict__ alpha_arr,
                                float* __restrict__ z_out,
                                float* __restrict__ s_out,
                                float* __restrict__ logits) {
    __shared__ float It[2][F_N * ROWPAD];   // double-buffered input tile
    __shared__ float zt[F_N * ROWPAD];
    __shared__ float st[F_N * ROWPAD];

    const int tid = threadIdx.x;   // 0..63 == filter index
    const int b   = blockIdx.x;

    const float alpha = alpha_arr[tid];
    const float omal  = 1.0f - alpha;
    float v = 0.0f;

    const float* Ib = Iin   + (size_t)b * F_N * L_LEN;
    float*       zb = z_out + (size_t)b * F_N * L_LEN;
    float*       sb = s_out + (size_t)b * F_N * L_LEN;

    // Prologue: async-fill buffer 0 with chunk 0.
    async_tile_issue(Ib, It[0], tid, 0);

    for (int c = 0; c < NCHUNK; ++c) {
        const int t0  = c * TS;
        const int buf = c & 1;

        // Current buffer's async copies complete, then make visible block-wide.
        asm volatile("s_wait_asynccnt 0x0" ::: "memory");
        __syncthreads();

        // Overlap: kick off the next chunk's copy into the other buffer.
        if (c + 1 < NCHUNK)
            async_tile_issue(Ib, It[buf ^ 1], tid, t0 + TS);

        // Sequential LIF scan: TS steps for this thread's filter row.
        {
            const float* Irow = &It[buf][tid * ROWPAD];
            float*       zrow = &zt[tid * ROWPAD];
            float*       srow = &st[tid * ROWPAD];
            #pragma unroll 8
            for (int tt = 0; tt < TS; ++tt) {
                float vp = alpha * v + omal * Irow[tt];
                float zz = BETA * (vp - THR);
                float sp = (vp >= THR) ? 1.0f : 0.0f;
                v = vp * (1.0f - sp);     // reset where spiked
                zrow[tt] = zz;
                srow[tt] = sp;
            }
        }
        __syncthreads();

        // Fused logits: column max over the 64 filters (conflict-free reads).
        {
            float m = -INFINITY;
            #pragma unroll 8
            for (int fr = 0; fr < F_N; ++fr)
                m = fmaxf(m, zt[fr * ROWPAD + tid]);
            logits[(size_t)b * L_LEN + t0 + tid] = m;
        }

        // Cooperative coalesced stores of z and s tiles.
        #pragma unroll
        for (int i = 0; i < 16; ++i) {
            int e  = i * 64 + tid;
            int fr = e >> 4;
            int tc = (e & 15) * 4;
            const float* zp = &zt[fr * ROWPAD + tc];
            const float* sp = &st[fr * ROWPAD + tc];
            float4 zv, sv;
            zv.x = zp[0]; zv.y = zp[1]; zv.z = zp[2]; zv.w = zp[3];
            sv.x = sp[0]; sv.y = sp[1]; sv.z = sp[2]; sv.w = sp[3];
            *reinterpret_cast<float4*>(zb + (size_t)fr * L_LEN + t0 + tc) = zv;
            *reinterpret_cast<float4*>(sb + (size_t)fr * L_LEN + t0 + tc) = sv;
        }
        // No trailing barrier: the wait+barrier at the top of the next
        // iteration orders the zt/st reads above against the next compute,
        // and the async fill targets the other It buffer (no alias).
    }
}

// ---------------------------------------------------------------------------
// Launch
// ---------------------------------------------------------------------------
extern "C" void kernel_launch(void* const* d_in, const int* in_sizes, int n_in,
                              void* d_out, int out_size, void* d_ws, size_t ws_size,
                              hipStream_t stream) {
    (void)in_sizes; (void)n_in; (void)out_size; (void)ws_size;

    const float* x       = (const float*)d_in[0];   // (32,1,8192)
    const float* conv_w  = (const float*)d_in[1];   // (64,1,8)
    const float* raw_tau = (const float*)d_in[2];   // (64,)

    float* out    = (float*)d_out;
    const size_t NE = (size_t)B_N * F_N * L_LEN;    // 16,777,216
    float* I      = out;                            // (32,64,8192)
    float* z      = out + NE;                       // (32,64,8192)
    float* s      = out + 2 * NE;                   // (32,64,8192)
    float* logit  = out + 3 * NE;                   // (32,8192)

    float* wn    = (float*)d_ws;                    // 64*8 floats
    float* alpha = wn + F_N * K_TAPS;               // 64 floats

    snn_prep_kernel<<<1, 64, 0, stream>>>(conv_w, raw_tau, wn, alpha);

    // 512 l-tiles / LT_PER_WAVE per wave -> grid.x = 64; 4 f-tiles; 32 batches.
    snn_conv_wmma_kernel<<<dim3(512 / LT_PER_WAVE, 4, B_N), 32, 0, stream>>>(x, wn, I);

    snn_scan_kernel<<<B_N, 64, 0, stream>>>(I, alpha, z, s, logit);
}